// SGDN_58497454571674
// MI455X (gfx1250) — compile-verified
//
#include <hip/hip_runtime.h>

// Problem constants (from reference)
#define U_N   100000
#define I_N   50000
#define R_N   5
#define E_N   100000
#define K_N   4
#define D_N   16
#define RD_N  64
#define OUT_N 64
#define KF_N  2
#define INV_TAU 2.0f   // 1/TAU, TAU = 0.5

#define TILES_PER_R   (E_N / 16)            // 6250
#define WAVES_PER_BLK 8                     // 256 threads / wave32
#define BLKS_PER_R    ((TILES_PER_R + WAVES_PER_BLK - 1) / WAVES_PER_BLK)  // 782

typedef float v2f __attribute__((ext_vector_type(2)));
typedef float v8f __attribute__((ext_vector_type(8)));

// ---------------------------------------------------------------------------
// Phase 1: per-edge attention weight (ead) + segment norms via f32 atomics
// ---------------------------------------------------------------------------
__global__ void sgdn_phase1_ead(const float* __restrict__ user_h,
                                const float* __restrict__ item_h,
                                const float* __restrict__ user_hsum,
                                const float* __restrict__ item_hsum,
                                const float* __restrict__ review_feat,
                                const float* __restrict__ prototypes,
                                const float* __restrict__ eta,
                                const int*   __restrict__ rows,
                                const int*   __restrict__ cols,
                                float* __restrict__ ead_ws,
                                float* __restrict__ norm_u,
                                float* __restrict__ norm_i)
{
    __shared__ __align__(16) float sproto[K_N * RD_N];   // 256 floats
    if (threadIdx.x < K_N * RD_N) sproto[threadIdx.x] = prototypes[threadIdx.x];
    __syncthreads();

    const long long gid = (long long)blockIdx.x * blockDim.x + threadIdx.x;
    if (gid >= (long long)R_N * E_N) return;
    const int r = (int)(gid / E_N);
    const size_t eoff = (size_t)gid;

    const int ro = rows[eoff];
    const int co = cols[eoff];

    // start the 1 KB streaming read early (global_prefetch_b8)
    const float* rfbase = review_feat + eoff * (K_N * RD_N);
    __builtin_prefetch(rfbase, 0, 1);
    __builtin_prefetch(rfbase + 128, 0, 1);

    // cosine similarity between user_h[r][ro] and item_h[r][co]
    const float4* uh = (const float4*)(user_h + ((size_t)r * U_N + ro) * D_N);
    const float4* ih = (const float4*)(item_h + ((size_t)r * I_N + co) * D_N);
    float dot = 0.f, nu = 0.f, ni = 0.f;
#pragma unroll
    for (int i = 0; i < 4; ++i) {
        float4 a = uh[i], b = ih[i];
        dot += a.x * b.x + a.y * b.y + a.z * b.z + a.w * b.w;
        nu  += a.x * a.x + a.y * a.y + a.z * a.z + a.w * a.w;
        ni  += b.x * b.x + b.y * b.y + b.z * b.z + b.w * b.w;
    }
    const float sim_k = dot * rsqrtf(nu * ni) * INV_TAU;

    // softmax denominator over K hsum channels
    const float4* us = (const float4*)(user_hsum + ((size_t)r * U_N + ro) * K_N * D_N);
    const float4* vs = (const float4*)(item_hsum + ((size_t)r * I_N + co) * K_N * D_N);
    float denom_sim = 0.f;
#pragma unroll
    for (int k = 0; k < K_N; ++k) {
        float d2 = 0.f;
#pragma unroll
        for (int i = 0; i < 4; ++i) {
            float4 a = us[k * 4 + i], b = vs[k * 4 + i];
            d2 += a.x * b.x + a.y * b.y + a.z * b.z + a.w * b.w;
        }
        denom_sim += __expf(d2 * INV_TAU);
    }
    const float exp_sim = __expf(sim_k) / denom_sim;

    // prototype attention over review features (dominant 1KB/edge read)
    const float4* rf = (const float4*)rfbase;
    const float4* pp = (const float4*)sproto;
    float ad_kf = 0.f, denom_ad = 0.f;
#pragma unroll
    for (int k = 0; k < K_N; ++k) {
        float d2 = 0.f;
#pragma unroll
        for (int i = 0; i < RD_N / 4; ++i) {
            float4 a = rf[k * (RD_N / 4) + i], p = pp[k * (RD_N / 4) + i];
            d2 += a.x * p.x + a.y * p.y + a.z * p.z + a.w * p.w;
        }
        const float adk = d2 * INV_TAU;
        if (k == KF_N) ad_kf = adk;
        denom_ad += __expf(adk);
    }
    const float ead0 = __expf(ad_kf) / denom_ad;

    const float g  = 1.f / (1.f + __expf(-eta[eoff]));
    const float ev = g * ead0 + (1.f - g) * exp_sim;

    ead_ws[eoff] = ev;
    atomicAdd(&norm_u[ro], ev);
    atomicAdd(&norm_i[co], ev);
}

// Scale C rows by per-edge weight and scatter with atomics.
// C layout: VGPR v <-> M = v + 8*half, column N = lane&15.
__device__ __forceinline__ void sgdn_scatter_scaled(v8f c, float wv, int segIdx,
                                                    int n, int half,
                                                    float* __restrict__ acc)
{
#pragma unroll
    for (int v = 0; v < 8; ++v) {
        const int M  = v + 8 * half;
        const float wM = __shfl(wv, M, 32);
        const int   sM = __shfl(segIdx, M, 32);
        atomicAdd(acc + (size_t)sM * D_N + n, c[v] * wM);
    }
}

// ---------------------------------------------------------------------------
// Phase 2: one wave per 16-edge tile, one relation per block.
// B fragments staged once per block into LDS in per-lane WMMA layout:
//   sfrag[dir][t][lane] = { W[k][n], W[k+1][n] },  k = 4*t' + 2*(lane>>4)
// Fused 16x80x16 chain: [node_row(16) | rf_k(64)] @ [node_w ; review_w],
// both directions interleaved so rf A-fragments are loaded once.
// ---------------------------------------------------------------------------
__global__ void sgdn_phase2_msg(const float* __restrict__ user_h,
                                const float* __restrict__ item_h,
                                const float* __restrict__ review_feat,
                                const float* __restrict__ node_w_fwd,
                                const float* __restrict__ node_w_rev,
                                const float* __restrict__ review_w_fwd,
                                const float* __restrict__ review_w_rev,
                                const int*   __restrict__ rows,
                                const int*   __restrict__ cols,
                                const float* __restrict__ ead_ws,
                                const float* __restrict__ norm_u,
                                const float* __restrict__ norm_i,
                                float* __restrict__ ufeat_acc,
                                float* __restrict__ ifeat_acc,
                                float* __restrict__ int_dist)
{
    __shared__ float2 sfrag[2][20][32];       // 10 KB of 320 KB/WGP LDS

    const int r  = blockIdx.x / BLKS_PER_R;   // uniform per block
    const int tb = blockIdx.x % BLKS_PER_R;

    // cooperative stage of both directions' B fragments
    for (int j = threadIdx.x; j < 2 * 20 * 32; j += blockDim.x) {
        const int dir = j / (20 * 32);
        const int rem = j % (20 * 32);
        const int t   = rem / 32;
        const int l   = rem % 32;
        const int h   = l >> 4;
        const int nn  = l & 15;
        const float* W;
        int k;
        if (t < 4) {
            k = 4 * t + 2 * h;
            W = (dir ? node_w_rev : node_w_fwd) + (size_t)r * D_N * D_N;
        } else {
            k = 4 * (t - 4) + 2 * h;
            W = (dir ? review_w_rev : review_w_fwd) + (size_t)r * RD_N * D_N;
        }
        sfrag[dir][t][l] = make_float2(W[k * D_N + nn], W[(k + 1) * D_N + nn]);
    }
    __syncthreads();

    const int waveInBlk = threadIdx.x >> 5;
    const int lane = threadIdx.x & 31;
    const int tile = tb * WAVES_PER_BLK + waveInBlk;
    if (tile >= TILES_PER_R) return;          // wave-uniform exit

    const int e0   = tile * 16;
    const int n    = lane & 15;               // A-row index == B/C column index
    const int half = lane >> 4;

    const size_t eoff = (size_t)r * E_N + (e0 + n);
    const int ro = rows[eoff];
    const int co = cols[eoff];
    const float wv = ead_ws[eoff] * rsqrtf(norm_u[ro] * norm_i[co]);
    if (half == 0) int_dist[eoff] = wv;

    const float* rfRow = review_feat + (eoff * K_N + KF_N) * RD_N;
    const float* uRow  = user_h + ((size_t)r * U_N + ro) * D_N;
    const float* iRow  = item_h + ((size_t)r * I_N + co) * D_N;

    v8f cf = {0.f, 0.f, 0.f, 0.f, 0.f, 0.f, 0.f, 0.f};
    v8f cr = {0.f, 0.f, 0.f, 0.f, 0.f, 0.f, 0.f, 0.f};

    // K = 0..15 : node projections (distinct A per direction)
#pragma unroll
    for (int t = 0; t < 4; ++t) {
        const int kc = 4 * t + 2 * half;
        v2f au = { uRow[kc], uRow[kc + 1] };
        const float2 bf = sfrag[0][t][lane];
        v2f vbf = { bf.x, bf.y };
        cf = __builtin_amdgcn_wmma_f32_16x16x4_f32(false, au, false, vbf, (short)0,
                                                   cf, false, false);
        v2f ai = { iRow[kc], iRow[kc + 1] };
        const float2 br = sfrag[1][t][lane];
        v2f vbr = { br.x, br.y };
        cr = __builtin_amdgcn_wmma_f32_16x16x4_f32(false, ai, false, vbr, (short)0,
                                                   cr, false, false);
    }
    // K = 16..79 : review projection (shared A fragment, two accumulators)
#pragma unroll
    for (int t = 4; t < 20; ++t) {
        const int kc = 4 * (t - 4) + 2 * half;
        v2f a = { rfRow[kc], rfRow[kc + 1] };
        const float2 bf = sfrag[0][t][lane];
        v2f vbf = { bf.x, bf.y };
        cf = __builtin_amdgcn_wmma_f32_16x16x4_f32(false, a, false, vbf, (short)0,
                                                   cf, false, false);
        const float2 br = sfrag[1][t][lane];
        v2f vbr = { br.x, br.y };
        cr = __builtin_amdgcn_wmma_f32_16x16x4_f32(false, a, false, vbr, (short)0,
                                                   cr, false, false);
    }

    sgdn_scatter_scaled(cf, wv, co, n, half, ifeat_acc);   // forward -> items
    sgdn_scatter_scaled(cr, wv, ro, n, half, ufeat_acc);   // reverse -> users
}

// ---------------------------------------------------------------------------
// Phase 3: leaky-ReLU + FC via WMMA (M x 16) @ (16 x 64) + bias.
// fc_w staged per block in per-lane fragment layout.
// ---------------------------------------------------------------------------
__global__ void sgdn_phase3_fc(const float* __restrict__ feat_acc,
                               const float* __restrict__ fc_w,
                               const float* __restrict__ fc_b,
                               float* __restrict__ out, int nRows)
{
    __shared__ float2 sfc[4][4][32];          // [ct][s][lane], 4 KB
    __shared__ float  sbias[OUT_N];

    for (int j = threadIdx.x; j < 4 * 4 * 32; j += blockDim.x) {
        const int ct = j / 128;
        const int s  = (j % 128) / 32;
        const int l  = j % 32;
        const int k  = 4 * s + 2 * (l >> 4);
        const int nn = l & 15;
        sfc[ct][s][l] = make_float2(fc_w[k * OUT_N + ct * 16 + nn],
                                    fc_w[(k + 1) * OUT_N + ct * 16 + nn]);
    }
    if (threadIdx.x < OUT_N) sbias[threadIdx.x] = fc_b[threadIdx.x];
    __syncthreads();

    const int wave = (int)((blockIdx.x * blockDim.x + threadIdx.x) >> 5);
    const int lane = threadIdx.x & 31;
    const int rowTiles = nRows / 16;
    if (wave >= rowTiles * (OUT_N / 16)) return;  // wave-uniform exit

    const int rt = wave >> 2;          // row tile
    const int ct = wave & 3;           // 16-wide column tile of OUT
    const int n    = lane & 15;
    const int half = lane >> 4;

    const float* arow = feat_acc + (size_t)(rt * 16 + n) * D_N;
    v8f c = {0.f, 0.f, 0.f, 0.f, 0.f, 0.f, 0.f, 0.f};
#pragma unroll
    for (int s = 0; s < 4; ++s) {
        const int kc = 4 * s + 2 * half;
        float a0 = arow[kc], a1 = arow[kc + 1];
        a0 = (a0 >= 0.f) ? a0 : 0.1f * a0;        // leaky ReLU on A
        a1 = (a1 >= 0.f) ? a1 : 0.1f * a1;
        v2f a = { a0, a1 };
        const float2 bb = sfc[ct][s][lane];
        v2f vb = { bb.x, bb.y };
        c = __builtin_amdgcn_wmma_f32_16x16x4_f32(false, a, false, vb, (short)0, c,
                                                  false, false);
    }
    const float bias = sbias[ct * 16 + n];
#pragma unroll
    for (int v = 0; v < 8; ++v) {
        const int M = v + 8 * half;
        out[(size_t)(rt * 16 + M) * OUT_N + ct * 16 + n] = c[v] + bias;
    }
}

// ---------------------------------------------------------------------------
extern "C" void kernel_launch(void* const* d_in, const int* in_sizes, int n_in,
                              void* d_out, int out_size, void* d_ws, size_t ws_size,
                              hipStream_t stream)
{
    const float* user_h       = (const float*)d_in[0];
    const float* item_h       = (const float*)d_in[1];
    const float* user_hsum    = (const float*)d_in[2];
    const float* item_hsum    = (const float*)d_in[3];
    const float* review_feat  = (const float*)d_in[4];
    const float* prototypes   = (const float*)d_in[5];
    const float* eta          = (const float*)d_in[6];
    const float* node_w_fwd   = (const float*)d_in[7];
    const float* node_w_rev   = (const float*)d_in[8];
    const float* review_w_fwd = (const float*)d_in[9];
    const float* review_w_rev = (const float*)d_in[10];
    const float* ufc_w        = (const float*)d_in[11];
    const float* ufc_b        = (const float*)d_in[12];
    const float* ifc_w        = (const float*)d_in[13];
    const float* ifc_b        = (const float*)d_in[14];
    const int*   rows         = (const int*)d_in[15];
    const int*   cols         = (const int*)d_in[16];

    float* out      = (float*)d_out;
    float* ufeat_o  = out;                             // U x 64
    float* ifeat_o  = out + (size_t)U_N * OUT_N;       // I x 64
    float* int_dist = ifeat_o + (size_t)I_N * OUT_N;   // R*E

    // workspace layout (contiguous)
    float* ead_ws    = (float*)d_ws;                   // R*E
    float* norm_u    = ead_ws + (size_t)R_N * E_N;     // U
    float* norm_i    = norm_u + U_N;                   // I
    float* ufeat_acc = norm_i + I_N;                   // U*D
    float* ifeat_acc = ufeat_acc + (size_t)U_N * D_N;  // I*D

    // zero the accumulators (norm_u..ifeat_acc are contiguous)
    hipMemsetAsync(norm_u, 0,
                   sizeof(float) * ((size_t)U_N + I_N +
                                    (size_t)U_N * D_N + (size_t)I_N * D_N),
                   stream);

    // Phase 1: edge attention weights + segment norms
    {
        const long long total = (long long)R_N * E_N;
        const int blocks = (int)((total + 255) / 256);
        sgdn_phase1_ead<<<blocks, 256, 0, stream>>>(
            user_h, item_h, user_hsum, item_hsum, review_feat, prototypes, eta,
            rows, cols, ead_ws, norm_u, norm_i);
    }

    // Phase 2: WMMA message passing (one wave per 16-edge tile, one r per block)
    {
        const int blocks = R_N * BLKS_PER_R;
        sgdn_phase2_msg<<<blocks, 256, 0, stream>>>(
            user_h, item_h, review_feat, node_w_fwd, node_w_rev,
            review_w_fwd, review_w_rev, rows, cols,
            ead_ws, norm_u, norm_i, ufeat_acc, ifeat_acc, int_dist);
    }

    // Phase 3: leaky + FC for users and items
    {
        const int wavesU = (U_N / 16) * (OUT_N / 16);
        sgdn_phase3_fc<<<(wavesU * 32 + 255) / 256, 256, 0, stream>>>(
            ufeat_acc, ufc_w, ufc_b, ufeat_o, U_N);
        const int wavesI = (I_N / 16) * (OUT_N / 16);
        sgdn_phase3_fc<<<(wavesI * 32 + 255) / 256, 256, 0, stream>>>(
            ifeat_acc, ifc_w, ifc_b, ifeat_o, I_N);
    }
}